// Model_1735166788321
// MI455X (gfx1250) — compile-verified
//
#include <hip/hip_runtime.h>
#include <math.h>

typedef float v2f __attribute__((ext_vector_type(2)));
typedef float v8f __attribute__((ext_vector_type(8)));
typedef unsigned int v4u __attribute__((ext_vector_type(4)));
typedef int v4i __attribute__((ext_vector_type(4)));
typedef int v8i __attribute__((ext_vector_type(8)));

#define WMMA_F32x4(a, b, c) \
  __builtin_amdgcn_wmma_f32_16x16x4_f32(false, (a), false, (b), (short)0, (c), false, false)

constexpr int Tlen = 4096;   // sequence length
constexpr int H    = 32;     // heads
constexpr int Pdim = 64;     // head dim p
constexpr int Ndim = 128;    // state dim n
constexpr int LCH  = 64;     // chunk length
constexpr int NCH  = 64;     // number of chunks (T / LCH)
constexpr int Bsz  = 2;      // batch

constexpr int LDN = Ndim + 4;  // padded LDS stride for 128-wide rows (TDM pad: 4 DWORDs)
constexpr int LDP = Pdim + 4;  // padded LDS stride for 64-wide rows

// ---------------------------------------------------------------------------
// Tensor Data Mover helpers (CDNA5 D#, CDNA5 ISA)
// 2-D tile: rows x cols f32 elements, global row stride in elements.
// pad codes: interval k -> 2^(k+1) DWORDs between pads, amount m -> m+1 DWORDs.
// pad_amount_code < 0 disables LDS padding.
// ---------------------------------------------------------------------------
__device__ __forceinline__ void tdm_desc(v4u& g0, v8i& g1, unsigned lds_addr,
                                         const void* gptr, unsigned rows,
                                         unsigned cols, unsigned stride_elems,
                                         int pad_interval_code, int pad_amount_code)
{
  unsigned long long ga = (unsigned long long)(uintptr_t)gptr;
  g0[0] = 1u;                                              // count=1, user mode
  g0[1] = lds_addr;                                        // lds_addr[31:0]
  g0[2] = (unsigned)ga;                                    // global_addr[31:0]
  g0[3] = (unsigned)((ga >> 32) & 0x01FFFFFFull) | (2u << 30);  // addr[56:32], type=2
  unsigned w0 = 2u << 16;                                  // data_size = 4 bytes
  if (pad_amount_code >= 0)
    w0 |= (1u << 20) | ((unsigned)pad_interval_code << 22) |
          ((unsigned)pad_amount_code << 25);
  g1[0] = (int)w0;                                         // mask=0, pads
  g1[1] = (int)(cols << 16);                               // tensor_dim0[15:0] @ bits63:48
  g1[2] = (int)((cols >> 16) | (rows << 16));              // dim0 hi | tensor_dim1 lo
  g1[3] = (int)((rows >> 16) | (cols << 16));              // dim1 hi | tile_dim0
  g1[4] = (int)(rows & 0xFFFFu);                           // tile_dim1, tile_dim2=0
  g1[5] = (int)stride_elems;                               // tensor_dim0_stride lo32
  g1[6] = 0;                                               // stride hi | dim1_stride lo
  g1[7] = 0;
}

__device__ __forceinline__ void tdm_load_2d(void* lds_ptr, const void* gptr,
                                            unsigned rows, unsigned cols,
                                            unsigned stride_elems,
                                            int pad_i, int pad_a)
{
  v4u g0; v8i g1;
  tdm_desc(g0, g1, (unsigned)(uintptr_t)lds_ptr, gptr, rows, cols, stride_elems,
           pad_i, pad_a);
  v4i z4 = {0, 0, 0, 0};
  v8i z8 = {0, 0, 0, 0, 0, 0, 0, 0};
  __builtin_amdgcn_tensor_load_to_lds(g0, g1, z4, z4, z8, 0);
}

__device__ __forceinline__ void tdm_store_2d(const void* lds_ptr, void* gptr,
                                             unsigned rows, unsigned cols,
                                             unsigned stride_elems)
{
  v4u g0; v8i g1;
  tdm_desc(g0, g1, (unsigned)(uintptr_t)lds_ptr, gptr, rows, cols, stride_elems,
           0, -1);
  v4i z4 = {0, 0, 0, 0};
  v8i z8 = {0, 0, 0, 0, 0, 0, 0, 0};
  __builtin_amdgcn_tensor_store_from_lds(g0, g1, z4, z4, z8, 0);
}

// ---------------------------------------------------------------------------
// Kernel 1: per (b, chunk, h):  cst[p, n] = sum_l exp(cs[63]-cs[l]) X[l,p] B[l,n]
// ---------------------------------------------------------------------------
__global__ __launch_bounds__(256) void k_chunk_states(
    const float* __restrict__ X, const float* __restrict__ A,
    const float* __restrict__ B, float* __restrict__ states,
    float* __restrict__ Alast)
{
  __shared__ float Xs[LCH][LDP];
  __shared__ float Bs[LCH][LDN];
  __shared__ float Cst[Pdim][Ndim];      // unpadded staging for TDM store
  __shared__ float cs[LCH];
  __shared__ float dv[LCH];

  const int bid = blockIdx.x;
  const int cI = bid & 63;
  const int h  = (bid >> 6) & 31;
  const int bI = bid >> 11;
  const int tid = threadIdx.x;
  const int wave = tid >> 5;
  const int lane = tid & 31;

  const float* Xg = X + ((size_t)(bI * Tlen + cI * LCH) * H + h) * Pdim;
  const float* Bg = B + ((size_t)(bI * Tlen + cI * LCH) * H + h) * Ndim;
  float* outp = states + ((size_t)((bI * H + h) * NCH + cI)) * (Pdim * Ndim);

  if (wave == 0) {                        // one TDM issue per block
    tdm_load_2d(&Xs[0][0], Xg, LCH, Pdim, H * Pdim, /*64dw*/5, /*4dw*/3);
    tdm_load_2d(&Bs[0][0], Bg, LCH, Ndim, H * Ndim, /*128dw*/6, /*4dw*/3);
  }
  if (tid < LCH)
    cs[tid] = A[(size_t)(bI * Tlen + cI * LCH + tid) * H + h];
  if (wave == 0) __builtin_amdgcn_s_wait_tensorcnt(0);
  __syncthreads();

  if (tid == 0) {                         // inclusive cumsum (64 adds, serial)
    float run = 0.f;
    for (int i = 0; i < LCH; i++) { run += cs[i]; cs[i] = run; }
  }
  __syncthreads();
  if (tid < LCH) dv[tid] = __expf(cs[LCH - 1] - cs[tid]);
  if (tid == 0) Alast[(bI * H + h) * NCH + cI] = cs[LCH - 1];
  __syncthreads();
  #pragma unroll
  for (int k = 0; k < 16; k++) {          // pre-scale X rows by decay once
    int idx = tid + 256 * k;
    int r = idx >> 6, c2 = idx & 63;
    Xs[r][c2] *= dv[r];
  }
  __syncthreads();

  const int nl = lane & 15;               // M/N index within fragment
  const int kb = (lane >> 4) << 1;        // K base: 0 or 2
  const int mb = (lane >> 4) << 3;        // C/D row base: 0 or 8

  // 32 output tiles: 4 p-tiles x 8 n-tiles; 4 per wave
  for (int t = wave; t < 32; t += 8) {
    const int i0 = (t >> 3) << 4;         // p tile base
    const int j0 = (t & 7) << 4;          // n tile base
    v8f acc = {};
    for (int k0 = 0; k0 < LCH; k0 += 4) {
      const int kk = k0 + kb;
      v2f a;  a.x = Xs[kk][i0 + nl];      // A[m,k] = (d*X)[l, p0+m]
              a.y = Xs[kk + 1][i0 + nl];
      v2f bb; bb.x = Bs[kk][j0 + nl];     // B[k,n'] = B[l, n0+n']
              bb.y = Bs[kk + 1][j0 + nl];
      acc = WMMA_F32x4(a, bb, acc);
    }
    #pragma unroll
    for (int v = 0; v < 8; v++)
      Cst[i0 + mb + v][j0 + nl] = acc[v];
  }
  __syncthreads();
  if (wave == 0) {                        // one DMA store of the 64x128 tile
    tdm_store_2d(&Cst[0][0], outp, Pdim, Ndim, Ndim);
    __builtin_amdgcn_s_wait_tensorcnt(0);
  }
}

// ---------------------------------------------------------------------------
// Kernel 2: per (b,h) sequential chunk scan, in place:
//   out[z] = S_z;  S_{z+1} = exp(A_last[z])*S_z + cst[z]
// ---------------------------------------------------------------------------
__global__ __launch_bounds__(256) void k_scan(
    float* __restrict__ states, const float* __restrict__ Alast)
{
  const int bh = blockIdx.x;              // 0..63
  const int tid = threadIdx.x;
  float4 S[8];
  #pragma unroll
  for (int k = 0; k < 8; k++) S[k] = make_float4(0.f, 0.f, 0.f, 0.f);

  for (int z = 0; z < NCH; z++) {
    const float e = __expf(Alast[bh * NCH + z]);
    float* p = states + ((size_t)bh * NCH + z) * (Pdim * Ndim);
    #pragma unroll
    for (int k = 0; k < 8; k++) {
      const int idx = (tid + 256 * k) * 4;
      float4 cst = *(const float4*)(p + idx);   // chunk-local state
      float4 sv = S[k];
      *(float4*)(p + idx) = sv;                 // overwrite with prefix state
      sv.x = e * sv.x + cst.x;  sv.y = e * sv.y + cst.y;
      sv.z = e * sv.z + cst.z;  sv.w = e * sv.w + cst.w;
      S[k] = sv;
    }
  }
}

// ---------------------------------------------------------------------------
// Kernel 3: per (b, chunk, h):
//   G = (C @ B^T) * L,  Y = G @ X + diag(exp(cs)) * (C @ S^T)
// ---------------------------------------------------------------------------
__global__ __launch_bounds__(256) void k_output(
    const float* __restrict__ X, const float* __restrict__ A,
    const float* __restrict__ B, const float* __restrict__ Cc,
    const float* __restrict__ states, float* __restrict__ Y)
{
  __shared__ float Csh[LCH][LDN];
  __shared__ float Bs [LCH][LDN];
  __shared__ float Ss [Pdim][LDN];
  __shared__ float Xs [LCH][LDP];
  __shared__ float G  [LCH][LDP];
  __shared__ float Yst[LCH][Pdim];        // unpadded staging for TDM store
  __shared__ float cs [LCH];

  const int bid = blockIdx.x;
  const int cI = bid & 63;
  const int h  = (bid >> 6) & 31;
  const int bI = bid >> 11;
  const int tid = threadIdx.x;
  const int wave = tid >> 5;
  const int lane = tid & 31;

  const float* Cg = Cc + ((size_t)(bI * Tlen + cI * LCH) * H + h) * Ndim;
  const float* Bg = B  + ((size_t)(bI * Tlen + cI * LCH) * H + h) * Ndim;
  const float* Sg = states + ((size_t)((bI * H + h) * NCH + cI)) * (Pdim * Ndim);
  const float* Xg = X + ((size_t)(bI * Tlen + cI * LCH) * H + h) * Pdim;
  float* Yg = Y + ((size_t)(bI * Tlen + cI * LCH) * H + h) * Pdim;

  if (wave == 0) {                        // 4 TDM tile loads for the block
    tdm_load_2d(&Csh[0][0], Cg, LCH,  Ndim, H * Ndim, 6, 3);
    tdm_load_2d(&Bs [0][0], Bg, LCH,  Ndim, H * Ndim, 6, 3);
    tdm_load_2d(&Ss [0][0], Sg, Pdim, Ndim, Ndim,     6, 3);
    tdm_load_2d(&Xs [0][0], Xg, LCH,  Pdim, H * Pdim, 5, 3);
  }
  if (tid < LCH)
    cs[tid] = A[(size_t)(bI * Tlen + cI * LCH + tid) * H + h];
  if (wave == 0) __builtin_amdgcn_s_wait_tensorcnt(0);
  __syncthreads();

  if (tid == 0) {
    float run = 0.f;
    for (int i = 0; i < LCH; i++) { run += cs[i]; cs[i] = run; }
  }
  __syncthreads();

  const int nl = lane & 15;
  const int kb = (lane >> 4) << 1;
  const int mb = (lane >> 4) << 3;

  // --- Step A: G = (C @ B^T) masked by L = exp(cs_i - cs_j), i >= j ---
  for (int t = wave; t < 16; t += 8) {
    const int i0 = (t >> 2) << 4;         // l-row tile
    const int j0 = (t & 3) << 4;          // s-col tile
    v8f acc = {};
    for (int k0 = 0; k0 < Ndim; k0 += 4) {
      const int kk = k0 + kb;
      v2f a;  a.x = Csh[i0 + nl][kk];  a.y = Csh[i0 + nl][kk + 1];  // A = C rows
      v2f bb; bb.x = Bs[j0 + nl][kk];  bb.y = Bs[j0 + nl][kk + 1];  // B = B^T
      acc = WMMA_F32x4(a, bb, acc);
    }
    #pragma unroll
    for (int v = 0; v < 8; v++) {
      const int r = i0 + mb + v, c2 = j0 + nl;
      G[r][c2] = (r >= c2) ? acc[v] * __expf(cs[r] - cs[c2]) : 0.f;
    }
  }
  __syncthreads();

  // --- Step B: Y = G @ X + exp(cs_row) * (C @ S^T) ---
  for (int t = wave; t < 16; t += 8) {
    const int i0 = (t >> 2) << 4;         // l tile
    const int j0 = (t & 3) << 4;          // p tile
    v8f accD = {};
    v8f accO = {};
    for (int k0 = 0; k0 < LCH; k0 += 4) {
      const int kk = k0 + kb;
      v2f a;  a.x = G[i0 + nl][kk];   a.y = G[i0 + nl][kk + 1];
      v2f bb; bb.x = Xs[kk][j0 + nl]; bb.y = Xs[kk + 1][j0 + nl];
      accD = WMMA_F32x4(a, bb, accD);
    }
    for (int k0 = 0; k0 < Ndim; k0 += 4) {
      const int kk = k0 + kb;
      v2f a;  a.x = Csh[i0 + nl][kk]; a.y = Csh[i0 + nl][kk + 1];
      v2f bb; bb.x = Ss[j0 + nl][kk]; bb.y = Ss[j0 + nl][kk + 1];   // B = S^T
      accO = WMMA_F32x4(a, bb, accO);
    }
    #pragma unroll
    for (int v = 0; v < 8; v++) {
      const int r = i0 + mb + v;
      Yst[r][j0 + nl] = accD[v] + __expf(cs[r]) * accO[v];
    }
  }
  __syncthreads();
  if (wave == 0) {                        // one DMA store of the 64x64 Y tile
    tdm_store_2d(&Yst[0][0], Yg, LCH, Pdim, H * Pdim);
    __builtin_amdgcn_s_wait_tensorcnt(0);
  }
}

// ---------------------------------------------------------------------------
extern "C" void kernel_launch(void* const* d_in, const int* in_sizes, int n_in,
                              void* d_out, int out_size, void* d_ws, size_t ws_size,
                              hipStream_t stream)
{
  const float* X = (const float*)d_in[0];
  const float* A = (const float*)d_in[1];
  const float* B = (const float*)d_in[2];
  const float* C = (const float*)d_in[3];
  float* Y = (float*)d_out;

  float* states = (float*)d_ws;                                  // 128 MiB
  float* Alast  = states + (size_t)Bsz * H * NCH * Pdim * Ndim;  // + 4096 f32

  dim3 blk(256);
  k_chunk_states<<<dim3(Bsz * H * NCH), blk, 0, stream>>>(X, A, B, states, Alast);
  k_scan<<<dim3(Bsz * H), blk, 0, stream>>>(states, Alast);
  k_output<<<dim3(Bsz * H * NCH), blk, 0, stream>>>(X, A, B, C, states, Y);
}